// EnvironmentEmbedder_2783138808252
// MI455X (gfx1250) — compile-verified
//
#include <hip/hip_runtime.h>

namespace {
constexpr int   kH       = 25;
constexpr int   kHW      = kH * kH;      // 625
constexpr int   kRot     = 6;
constexpr int   kEmb     = 128;
constexpr int   kOutC    = 161;          // 128 env + 26 aux + 1 ones + 6 compass
constexpr float kVScale  = 0.5f;
constexpr int   kThreads = 256;          // 8 wave32 per block
}

__global__ __launch_bounds__(kThreads)
void env_embed_kernel(const float* __restrict__ est,      // (B,128,25,25)
                      const float* __restrict__ edy,      // (B,128,25,25)
                      const float* __restrict__ obstacle, // (B,25,25)
                      const float* __restrict__ obs_cur,  // (B,25,25)
                      const float* __restrict__ obs_mem,  // (B,25,25)
                      const float* __restrict__ prev_vis, // (B,6,25,25)
                      const float* __restrict__ all_tgt,  // (B,6,25,25)
                      const float* __restrict__ prev_tgt, // (B,6,25,25)
                      const float* __restrict__ leader,   // (B,25,25)
                      const float* __restrict__ follower, // (B,25,25)
                      const int*   __restrict__ rot,      // (B,)
                      float* __restrict__ out)            // (B,161,25,25)
{
    __shared__ float s_m[kHW];   // obs_mem row, reused by channels 0..153

    const int b   = blockIdx.x;
    const int tid = threadIdx.x;

    // Stage obs_mem[b] in LDS (2.5 KB of the 320 KB WGP LDS).
    const float* __restrict__ mrow = obs_mem + (size_t)b * kHW;
    for (int p = tid; p < kHW; p += kThreads) s_m[p] = mrow[p];
    __syncthreads();

    const int r = rot[b];  // block-uniform -> SGPR

    const float* __restrict__ sb = est + (size_t)b * (kEmb * kHW);
    const float* __restrict__ db = edy + (size_t)b * (kEmb * kHW);
    float* __restrict__ ob = out + (size_t)b * ((size_t)kOutC * kHW);

    // ---------------- Bulk: channels 0..127, (static+dynamic)*m ----------------
    // Flat over 128*625 = 80000 floats = 20000 float4. Input batch bases are
    // 16B aligned; output batch base may be only 4B aligned -> memcpy store.
    constexpr int kBulk = kEmb * kHW;   // 80000
    constexpr int kV4   = kBulk / 4;    // 20000
    for (int v = tid; v < kV4; v += kThreads) {
        const int i = v * 4;
        __builtin_prefetch(sb + i + kThreads * 4, 0, 1);  // global_prefetch_b8
        __builtin_prefetch(db + i + kThreads * 4, 0, 1);
        const float4 s4 = *reinterpret_cast<const float4*>(sb + i);  // b128
        const float4 d4 = *reinterpret_cast<const float4*>(db + i);  // b128
        int p = i % kHW;                       // pixel index of element 0
        const float m0 = s_m[p]; p = (p + 1 == kHW) ? 0 : p + 1;
        const float m1 = s_m[p]; p = (p + 1 == kHW) ? 0 : p + 1;
        const float m2 = s_m[p]; p = (p + 1 == kHW) ? 0 : p + 1;
        const float m3 = s_m[p];
        float4 o;
        o.x = (s4.x + d4.x) * m0;
        o.y = (s4.y + d4.y) * m1;
        o.z = (s4.z + d4.z) * m2;
        o.w = (s4.w + d4.w) * m3;
        __builtin_memcpy(ob + i, &o, sizeof(float4));
    }

    // ---------------- Small channels 128..160 ----------------
    const size_t rb = (size_t)b * kRot * kHW;
    const float* __restrict__ pv = prev_vis + rb;
    const float* __restrict__ at = all_tgt  + rb;
    const float* __restrict__ pt = prev_tgt + rb;
    const float* __restrict__ obst = obstacle + (size_t)b * kHW;
    const float* __restrict__ ocur = obs_cur  + (size_t)b * kHW;
    const float* __restrict__ led  = leader   + (size_t)b * kHW;
    const float* __restrict__ fol  = follower + (size_t)b * kHW;

    for (int p = tid; p < kHW; p += kThreads) {
        const float m  = s_m[p];
        const float hm = kVScale * m;   // exact power-of-two scale

        float v[kRot], a[kRot], t[kRot];
        float sv = 0.f, sa = 0.f, sp = 0.f;
        #pragma unroll
        for (int k = 0; k < kRot; ++k) {
            v[k] = pv[k * kHW + p]; sv += v[k];
            a[k] = at[k * kHW + p]; sa += a[k];
            t[k] = pt[k * kHW + p]; sp += t[k];
        }

        ob[128 * kHW + p] = obst[p] * m;   // obstacle_mask
        ob[129 * kHW + p] = ocur[p] * m;   // observability_current
        ob[130 * kHW + p] = m * m;         // obs_mem channel * obs_mem

        // Egocentric shuffle applied on the store side: src k -> dst (k+r)%6.
        #pragma unroll
        for (int k = 0; k < kRot; ++k) {
            int j = k + r; if (j >= kRot) j -= kRot;
            ob[(131 + j) * kHW + p] = v[k] * hm;  // prev_vis_scaled * m
            ob[(140 + j) * kHW + p] = a[k] * hm;  // 0.5 * all_prev_tgt * m
            ob[(146 + j) * kHW + p] = t[k] * m;   // prev_tgt * m
        }

        ob[137 * kHW + p] = sv * m;        // all_visitations (unscaled sum)
        ob[138 * kHW + p] = led[p] * m;    // leader
        ob[139 * kHW + p] = fol[p] * m;    // follower
        ob[152 * kHW + p] = sa * hm;       // all_pos_prev (0.5 * sum)
        ob[153 * kHW + p] = sp * m;        // pos_prev
        ob[154 * kHW + p] = 1.0f;          // env_channel (NOT scaled by m)

        #pragma unroll
        for (int j = 0; j < kRot; ++j)     // compass one-hot (NOT scaled by m)
            ob[(155 + j) * kHW + p] = (j == r) ? 1.0f : 0.0f;
    }
}

extern "C" void kernel_launch(void* const* d_in, const int* in_sizes, int n_in,
                              void* d_out, int out_size, void* d_ws, size_t ws_size,
                              hipStream_t stream) {
    const int B = in_sizes[2] / kHW;   // obstacle_mask is (B, 25, 25)
    env_embed_kernel<<<B, kThreads, 0, stream>>>(
        (const float*)d_in[0],  // embedded_static
        (const float*)d_in[1],  // embedded_dynamic
        (const float*)d_in[2],  // obstacle_mask
        (const float*)d_in[3],  // observability_current
        (const float*)d_in[4],  // observability_in_memory
        (const float*)d_in[5],  // previous_visitations
        (const float*)d_in[6],  // all_previous_targets
        (const float*)d_in[7],  // previous_target
        (const float*)d_in[8],  // leader_location
        (const float*)d_in[9],  // follower_location
        (const int*)  d_in[10], // rotations
        (float*)d_out);
}